// K_max_pooling1d_9740985827591
// MI455X (gfx1250) — compile-verified
//
#include <hip/hip_runtime.h>
#include <stdint.h>

#define NB    32          // batch
#define NL    16384       // sequence length
#define NC    128         // channels
#define NK    64          // k of top-k
#define TPB   128         // threads per block (= NC), 4 wave32 waves
#define CHUNK 16          // rows per async chunk
#define CHUNK_BYTES (CHUNK * NC * 4)   // 8 KB

// ---------------- CDNA5 async-to-LDS primitives ----------------
// GLOBAL_LOAD_ASYNC_TO_LDS_B128 (GV mode):
//   copies 16 B per lane from global memory to LDS, counted by ASYNCcnt
// The immediate offset applies to BOTH addresses (ISA 08 §4.4), and our
// intra-chunk layout is identical on both sides, so one address pair serves
// four ops. Streaming input is read once -> non-temporal hint to spare L2;
// merge-phase candidate reads keep default policy so they hit L2.

#define ASYNC_B128_NT(lds, g, off)                                              \
  asm volatile("global_load_async_to_lds_b128 %0, %1, off offset:" #off         \
               " th:TH_LOAD_NT"                                                 \
               :: "v"(lds), "v"(g) : "memory")

#define ASYNC_B128(lds, g, off)                                                 \
  asm volatile("global_load_async_to_lds_b128 %0, %1, off offset:" #off         \
               :: "v"(lds), "v"(g) : "memory")

#if __has_builtin(__builtin_amdgcn_s_wait_asynccnt)
#define WAITA(n) __builtin_amdgcn_s_wait_asynccnt(n)
#else
#define WAITA(n) asm volatile("s_wait_asynccnt " #n ::: "memory")
#endif

// wait ASYNCcnt <= 4*k (kernel-1 units: 4 async ops per chunk)
__device__ __forceinline__ void wait_chunks4(int k) {
  switch (k) {
    case 0: WAITA(0);  break;
    case 1: WAITA(4);  break;
    case 2: WAITA(8);  break;
    case 3: WAITA(12); break;
    case 4: WAITA(16); break;
    case 5: WAITA(20); break;
    default: WAITA(24); break;
  }
}

// wait ASYNCcnt <= 8*k (merge units: 8 async ops per chunk — values + indices)
__device__ __forceinline__ void wait_chunks8(int k) {
  switch (k) {
    case 0: WAITA(0);  break;
    case 1: WAITA(8);  break;
    case 2: WAITA(16); break;
    case 3: WAITA(24); break;
    case 4: WAITA(32); break;
    case 5: WAITA(40); break;
    default: WAITA(48); break;
  }
}

// Per-lane byte offset so each wave's 4 B128 ops cover exactly the 16 rows x
// 128 B slice (channels [32w, 32w+32)) that its own lanes consume: no
// cross-wave deps, no workgroup barriers in the streaming loop.
__device__ __forceinline__ uint32_t lane_byte(int tid) {
  return (uint32_t)(((tid >> 5) * 128) + (((tid >> 3) & 3) * 512) + ((tid & 7) * 16));
}

// ---------------- per-thread top-k helpers (state lives in LDS) ----------------

// Locate the entry that would be evicted next: smallest value; among equal
// values, smallest index (larger index wins ties, matching jnp.argsort
// stable-ascending + take-last-K semantics).
__device__ __forceinline__ void rescan_min(const float* tv, const int* ti, int tid,
                                           float& mn, int& mnIdx, int& mnPos) {
  mn = tv[tid]; mnIdx = ti[tid]; mnPos = 0;
  for (int j = 1; j < NK; ++j) {
    float vv = tv[j * TPB + tid];
    int   ii = ti[j * TPB + tid];
    if (vv < mn || (vv == mn && ii < mnIdx)) { mn = vv; mnIdx = ii; mnPos = j; }
  }
}

__device__ __forceinline__ void topk_insert(float* tv, int* ti, int tid,
                                            float v, int l, int& count,
                                            float& mn, int& mnIdx, int& mnPos) {
  if (count < NK) {
    tv[count * TPB + tid] = v;
    ti[count * TPB + tid] = l;
    ++count;
    if (count == NK) rescan_min(tv, ti, tid, mn, mnIdx, mnPos);
  } else if (v > mn || (v == mn && l > mnIdx)) {
    tv[mnPos * TPB + tid] = v;
    ti[mnPos * TPB + tid] = l;
    rescan_min(tv, ti, tid, mn, mnIdx, mnPos);
  }
}

// Emit the 64 kept entries in ascending original-index order (selection sort).
__device__ __forceinline__ void emit_sorted(float* tv, int* ti, int tid,
                                            float* __restrict__ out, int b) {
  for (int j = 0; j < NK; ++j) {
    int bestPos = 0;
    int bestIdx = ti[tid];
    for (int q = 1; q < NK; ++q) {
      int ii = ti[q * TPB + tid];
      if (ii < bestIdx) { bestIdx = ii; bestPos = q; }
    }
    out[((size_t)b * NK + j) * NC + tid] = tv[bestPos * TPB + tid];
    ti[bestPos * TPB + tid] = 0x7fffffff;  // consume
  }
}

// ---------------- kernel 1: stream a segment, keep per-column top-64 ----------------

template <int SEG, bool FINAL>
__global__ void __launch_bounds__(TPB)
kmax_seg_kernel(const float* __restrict__ x,
                float* __restrict__ cand_v, int* __restrict__ cand_i,
                float* __restrict__ out) {
  constexpr int LSEG  = NL / SEG;
  constexpr int NCH   = LSEG / CHUNK;   // >= 64 for SEG <= 16
  constexpr int NBUF  = 8;              // 64 KB staging ring
  constexpr int AHEAD = 7;              // 7 chunks * 4 ops = 28 outstanding (cap 63)

  __shared__ float stage[NBUF][CHUNK * NC];  // 64 KB
  __shared__ float tv[NK * TPB];             // 32 KB
  __shared__ int   ti[NK * TPB];             // 32 KB

  const int tid = threadIdx.x;
  const int b   = blockIdx.x / SEG;          // SEG is pow2 -> shift
  const int seg = blockIdx.x % SEG;
  const int l0  = seg * LSEG;

  const uint32_t lb = lane_byte(tid);
  const uint64_t gbase =
      (uint64_t)(uintptr_t)(x + ((size_t)b * NL + l0) * NC) + lb;
  // Flat shared pointer truncated to 32 bits == hardware LDS offset (ISA §10.2).
  const uint32_t ldsbase = (uint32_t)(uintptr_t)&stage[0][0] + lb;

  auto issue = [&](int chunk) {
    uint64_t g = gbase + (uint64_t)chunk * CHUNK_BYTES;
    uint32_t l = ldsbase + (uint32_t)(chunk & (NBUF - 1)) * CHUNK_BYTES;
    ASYNC_B128_NT(l, g, 0);      // rows 0-3 of this wave's slice
    ASYNC_B128_NT(l, g, 2048);   // rows 4-7
    ASYNC_B128_NT(l, g, 4096);   // rows 8-11
    ASYNC_B128_NT(l, g, 6144);   // rows 12-15
  };

  int count = 0, mnIdx = 0, mnPos = 0;
  float mn = 0.0f;

  auto process = [&](int chunk) {
    const float* s = &stage[chunk & (NBUF - 1)][0];
    const int lbase = l0 + chunk * CHUNK;
    for (int r = 0; r < CHUNK; ++r) {
      float v = s[r * NC + tid];
      topk_insert(tv, ti, tid, v, lbase + r, count, mn, mnIdx, mnPos);
    }
  };

  for (int c = 0; c < AHEAD; ++c) issue(c);

  int i = 0;
  for (; i < NCH - AHEAD; ++i) {   // steady state: 6 younger chunks outstanding
    wait_chunks4(6);
    process(i);
    issue(i + AHEAD);
  }
  for (; i < NCH; ++i) {           // drain
    wait_chunks4(NCH - 1 - i);
    process(i);
  }

  if (FINAL) {
    emit_sorted(tv, ti, tid, out, b);
  } else {
    const size_t base = ((size_t)b * SEG + seg) * NK;
    for (int j = 0; j < NK; ++j) {
      cand_v[(base + j) * NC + tid] = tv[j * TPB + tid];
      cand_i[(base + j) * NC + tid] = ti[j * TPB + tid];
    }
  }
}

// ---------------- kernel 2: merge SEG candidate sets per column ----------------

template <int SEG>
__global__ void __launch_bounds__(TPB)
kmax_merge_kernel(const float* __restrict__ cand_v, const int* __restrict__ cand_i,
                  float* __restrict__ out) {
  constexpr int M     = SEG * NK;       // candidate rows per batch
  constexpr int NCH   = M / CHUNK;
  constexpr int NBUF  = 8;              // 2 * 64 KB staging rings
  constexpr int AHEAD = 7;              // 7 chunks * 8 ops = 56 outstanding (cap 63)

  __shared__ float sv[NBUF][CHUNK * NC];  // 64 KB values stage
  __shared__ int   si[NBUF][CHUNK * NC];  // 64 KB indices stage
  __shared__ float tv[NK * TPB];          // 32 KB
  __shared__ int   ti[NK * TPB];          // 32 KB

  const int tid = threadIdx.x;
  const int b   = blockIdx.x;

  const uint32_t lb = lane_byte(tid);
  const uint64_t gv = (uint64_t)(uintptr_t)(cand_v + (size_t)b * M * NC) + lb;
  const uint64_t gi = (uint64_t)(uintptr_t)(cand_i + (size_t)b * M * NC) + lb;
  const uint32_t lv = (uint32_t)(uintptr_t)&sv[0][0] + lb;
  const uint32_t li = (uint32_t)(uintptr_t)&si[0][0] + lb;

  auto issue = [&](int chunk) {
    uint64_t dgv = gv + (uint64_t)chunk * CHUNK_BYTES;
    uint64_t dgi = gi + (uint64_t)chunk * CHUNK_BYTES;
    uint32_t o   = (uint32_t)(chunk & (NBUF - 1)) * CHUNK_BYTES;
    uint32_t olv = lv + o;
    uint32_t oli = li + o;
    ASYNC_B128(olv, dgv, 0);      // candidate values: L2-resident, default policy
    ASYNC_B128(olv, dgv, 2048);
    ASYNC_B128(olv, dgv, 4096);
    ASYNC_B128(olv, dgv, 6144);
    ASYNC_B128(oli, dgi, 0);      // candidate indices
    ASYNC_B128(oli, dgi, 2048);
    ASYNC_B128(oli, dgi, 4096);
    ASYNC_B128(oli, dgi, 6144);
  };

  int count = 0, mnIdx = 0, mnPos = 0;
  float mn = 0.0f;

  auto process = [&](int chunk) {
    const float* pv = &sv[chunk & (NBUF - 1)][0];
    const int*   pi = &si[chunk & (NBUF - 1)][0];
    for (int r = 0; r < CHUNK; ++r) {
      float v = pv[r * NC + tid];
      int   l = pi[r * NC + tid];
      topk_insert(tv, ti, tid, v, l, count, mn, mnIdx, mnPos);
    }
  };

  constexpr int PRE = (AHEAD < NCH) ? AHEAD : NCH;
  for (int c = 0; c < PRE; ++c) issue(c);

  int i = 0;
  for (; i < NCH - AHEAD; ++i) {
    wait_chunks8(6);
    process(i);
    issue(i + AHEAD);
  }
  for (; i < NCH; ++i) {
    int yk = NCH - 1 - i;
    wait_chunks8(yk < 6 ? yk : 6);
    process(i);
  }

  emit_sorted(tv, ti, tid, out, b);
}

// ---------------- launch ----------------

extern "C" void kernel_launch(void* const* d_in, const int* in_sizes, int n_in,
                              void* d_out, int out_size, void* d_ws, size_t ws_size,
                              hipStream_t stream) {
  (void)in_sizes; (void)n_in; (void)out_size;
  const float* x = (const float*)d_in[0];
  float* out = (float*)d_out;

  // Candidate scratch: SEG * (B*K*C) * (4B val + 4B idx) = SEG * 2 MiB.
  const size_t per_seg = (size_t)NB * NK * NC * 8;
  int SEG = 16;
  while (SEG > 1 && (size_t)SEG * per_seg > ws_size) SEG >>= 1;

  float* cv = (float*)d_ws;
  int*   ci = (int*)((char*)d_ws + (size_t)SEG * per_seg / 2);

  switch (SEG) {
    case 16:
      kmax_seg_kernel<16, false><<<NB * 16, TPB, 0, stream>>>(x, cv, ci, out);
      kmax_merge_kernel<16><<<NB, TPB, 0, stream>>>(cv, ci, out);
      break;
    case 8:
      kmax_seg_kernel<8, false><<<NB * 8, TPB, 0, stream>>>(x, cv, ci, out);
      kmax_merge_kernel<8><<<NB, TPB, 0, stream>>>(cv, ci, out);
      break;
    case 4:
      kmax_seg_kernel<4, false><<<NB * 4, TPB, 0, stream>>>(x, cv, ci, out);
      kmax_merge_kernel<4><<<NB, TPB, 0, stream>>>(cv, ci, out);
      break;
    case 2:
      kmax_seg_kernel<2, false><<<NB * 2, TPB, 0, stream>>>(x, cv, ci, out);
      kmax_merge_kernel<2><<<NB, TPB, 0, stream>>>(cv, ci, out);
      break;
    default:
      // Degenerate workspace: single-pass, sort + write directly.
      kmax_seg_kernel<1, true><<<NB, TPB, 0, stream>>>(x, nullptr, nullptr, out);
      break;
  }
}